// ODEFunc_83915071029947
// MI455X (gfx1250) — compile-verified
//
#include <hip/hip_runtime.h>
#include <hip/hip_bf16.h>

#define N_NODES 100000
#define N_EDGES 3200000
#define PDIM 32
#define QDIM 32
#define DDIM 64

typedef __attribute__((ext_vector_type(2))) float v2f;
typedef __attribute__((ext_vector_type(8))) float v8f;

// ---------------------------------------------------------------------------
// Kernel 1: zero the workspace (nbr_sum[N][Q] followed by deg[N]).
// ---------------------------------------------------------------------------
__global__ void zero_ws_kernel(float* __restrict__ ws, int n) {
    int i = blockIdx.x * blockDim.x + threadIdx.x;
    if (i < n) ws[i] = 0.0f;
}

// ---------------------------------------------------------------------------
// Kernel 2: edge scatter. One wave32 per edge; lane q handles column q of Q=32.
// intensity gather + nbr_sum scatter are both L2-resident (12.8 MB), atomics
// are coalesced 128B per edge -> hardware global_atomic_add_f32.
// ---------------------------------------------------------------------------
__global__ void edge_scatter_kernel(const float* __restrict__ intensity,
                                    const int* __restrict__ edge_src,
                                    const int* __restrict__ edge_dst,
                                    float* __restrict__ nbr_sum,
                                    float* __restrict__ deg,
                                    int nE) {
    int gid  = blockIdx.x * blockDim.x + threadIdx.x;
    int e    = gid >> 5;
    int lane = threadIdx.x & 31;
    if (e >= nE) return;
    int s = edge_src[e];
    int d = edge_dst[e];
    float v = intensity[(size_t)s * QDIM + lane];
    unsafeAtomicAdd(&nbr_sum[(size_t)d * QDIM + lane], v);
    if (lane == 0) unsafeAtomicAdd(&deg[d], 1.0f);
}

// ---------------------------------------------------------------------------
// Fused node kernel. One wave = 16 node rows. All GEMMs via
// V_WMMA_F32_16X16X4_F32 (exact f32, D = A(16x4) x B(4x16) + C).
// ---------------------------------------------------------------------------
__device__ __forceinline__ v8f wmma_f32x4(v2f a, v2f b, v8f c) {
    // 8 args: (neg_a, A, neg_b, B, c_mod, C, reuse_a, reuse_b)
    return __builtin_amdgcn_wmma_f32_16x16x4_f32(
        false, a, false, b, (short)0, c, false, false);
}

__device__ __forceinline__ float softplus_f(float x) {
    // numerically stable log1p(exp(x))
    return fmaxf(x, 0.0f) + log1pf(__expf(-fabsf(x)));
}

__global__ __launch_bounds__(256) void fused_node_kernel(
    const float* __restrict__ u,
    const float* __restrict__ intensity,
    const float* __restrict__ nbr_sum,
    const float* __restrict__ deg,
    const float* __restrict__ Wf, const float* __restrict__ bf,
    const float* __restrict__ Wg, const float* __restrict__ bg,
    const float* __restrict__ Wz, const float* __restrict__ bz,
    const float* __restrict__ WA, const float* __restrict__ bA,
    float* __restrict__ out) {

    const int lane  = threadIdx.x & 31;
    const int wave  = threadIdx.x >> 5;
    const int base  = blockIdx.x * 128 + wave * 16;   // 8 waves x 16 rows
    if (base >= N_NODES) return;                      // wave-uniform: EXEC stays all-1s

    const int ln    = lane & 15;                 // A-row / B,C,D-column within tile
    const int koff  = (lane < 16) ? 0 : 2;       // K sub-offset for A/B fragments
    const int row   = base + ln;                 // A-matrix row this lane feeds
    const int chalf = (lane < 16) ? 0 : 8;       // C/D row-half offset

    // ---- A fragments for aggregation input: cols 0-31 intensity, 32-63 nbr_mean
    const float dg  = deg[row];
    const float inv = (dg > 0.0f) ? (1.0f / dg) : 0.0f;  // where(deg>0, 1/max(deg,1), 0)

    v2f aggA[16];
#pragma unroll
    for (int kb = 0; kb < 8; ++kb) {
        const float* p = intensity + (size_t)row * QDIM + 4 * kb + koff;
        v2f a; a.x = p[0]; a.y = p[1];
        aggA[kb] = a;
    }
#pragma unroll
    for (int kb = 0; kb < 8; ++kb) {
        const float* p = nbr_sum + (size_t)row * QDIM + 4 * kb + koff;
        v2f a; a.x = p[0] * inv; a.y = p[1] * inv;
        aggA[8 + kb] = a;
    }

    // ---- aggout = relu([intensity|nbr_mean] @ WA^T + bA)  (two 16-col tiles)
    v8f agg[2];
#pragma unroll
    for (int jt = 0; jt < 2; ++jt) {
        const int j0 = jt * 16;
        const float bias = bA[j0 + ln];
        v8f c;
#pragma unroll
        for (int r = 0; r < 8; ++r) c[r] = bias;
#pragma unroll
        for (int kb = 0; kb < 16; ++kb) {
            const float* wp = WA + (size_t)(j0 + ln) * DDIM + 4 * kb + koff;
            v2f b; b.x = wp[0]; b.y = wp[1];
            c = wmma_f32x4(aggA[kb], b, c);
        }
#pragma unroll
        for (int r = 0; r < 8; ++r) c[r] = fmaxf(c[r], 0.0f);
        agg[jt] = c;
    }

    // ---- A fragments for u (shared by Fu / Gu / Zu GEMMs)
    v2f uA[16];
#pragma unroll
    for (int kb = 0; kb < 16; ++kb) {
        const float* p = u + (size_t)row * DDIM + 4 * kb + koff;
        v2f a; a.x = p[0]; a.y = p[1];
        uA[kb] = a;
    }

    // generic 16-col-tile GEMM against u: W row-major [nout][64]
    auto gemm_u = [&](const float* __restrict__ W, const float* __restrict__ b,
                      int j0) -> v8f {
        const float bias = b[j0 + ln];
        v8f c;
#pragma unroll
        for (int r = 0; r < 8; ++r) c[r] = bias;
#pragma unroll
        for (int kb = 0; kb < 16; ++kb) {
            const float* wp = W + (size_t)(j0 + ln) * DDIM + 4 * kb + koff;
            v2f bb; bb.x = wp[0]; bb.y = wp[1];
            c = wmma_f32x4(uA[kb], bb, c);
        }
        return c;
    };

    // ---- Zu = tanh(u @ Wz^T + bz)   (two 16-col tiles)
    v8f zt[2];
#pragma unroll
    for (int jt = 0; jt < 2; ++jt) {
        v8f c = gemm_u(Wz, bz, jt * 16);
#pragma unroll
        for (int r = 0; r < 8; ++r) c[r] = tanhf(c[r]);
        zt[jt] = c;
    }

    // ---- per-column-tile: F = softplus, G = softplus, du = -F*u + G*[Z|agg]
    v8f dup[2], up[2];
#pragma unroll
    for (int jt = 0; jt < 4; ++jt) {
        const int j0 = jt * 16;
        v8f F = gemm_u(Wf, bf, j0);
        v8f G = gemm_u(Wg, bg, j0);
        v8f S = (jt < 2) ? zt[jt] : agg[jt - 2];

        v8f uc;
#pragma unroll
        for (int r = 0; r < 8; ++r)
            uc[r] = u[(size_t)(base + chalf + r) * DDIM + j0 + ln];

        v8f d;
#pragma unroll
        for (int r = 0; r < 8; ++r) {
            const float f = softplus_f(F[r]);
            const float g = softplus_f(G[r]);
            d[r] = -f * uc[r] + g * S[r];
        }

        if (jt < 2) {              // first P=32 columns: keep for projection
            dup[jt] = d;
            up[jt]  = uc;
        } else {                   // q-part: store directly
#pragma unroll
            for (int r = 0; r < 8; ++r)
                out[(size_t)(base + chalf + r) * DDIM + j0 + ln] = d[r];
        }
    }

    // ---- projection: dup -= (sum(dup*up)/sum(up*up)) * up   (row-wise over 32 cols)
    // Row m's 16 columns of a tile live in one 16-lane half at VGPR slot m&7,
    // so xor-shuffles with masks 1,2,4,8 reduce exactly one row.
#pragma unroll
    for (int r = 0; r < 8; ++r) {
        float pd = dup[0][r] * up[0][r] + dup[1][r] * up[1][r];
        float pn = up[0][r]  * up[0][r] + up[1][r]  * up[1][r];
#pragma unroll
        for (int m = 1; m < 16; m <<= 1) {
            pd += __shfl_xor(pd, m, 32);
            pn += __shfl_xor(pn, m, 32);
        }
        const float s = pd / pn;
        dup[0][r] -= s * up[0][r];
        dup[1][r] -= s * up[1][r];
    }

#pragma unroll
    for (int jt = 0; jt < 2; ++jt)
#pragma unroll
        for (int r = 0; r < 8; ++r)
            out[(size_t)(base + chalf + r) * DDIM + jt * 16 + ln] = dup[jt][r];
}

// ---------------------------------------------------------------------------
extern "C" void kernel_launch(void* const* d_in, const int* in_sizes, int n_in,
                              void* d_out, int out_size, void* d_ws, size_t ws_size,
                              hipStream_t stream) {
    const float* u         = (const float*)d_in[0];
    const float* intensity = (const float*)d_in[1];
    const int*   edge_src  = (const int*)d_in[2];
    const int*   edge_dst  = (const int*)d_in[3];
    const float* Wf        = (const float*)d_in[4];
    const float* bf        = (const float*)d_in[5];
    const float* Wg        = (const float*)d_in[6];
    const float* bg        = (const float*)d_in[7];
    const float* Wz        = (const float*)d_in[8];
    const float* bz        = (const float*)d_in[9];
    const float* WA        = (const float*)d_in[10];
    const float* bA        = (const float*)d_in[11];
    float* out = (float*)d_out;

    float* nbr_sum = (float*)d_ws;                      // N * Q floats
    float* deg     = nbr_sum + (size_t)N_NODES * QDIM;  // N floats

    const int nz = N_NODES * (QDIM + 1);
    zero_ws_kernel<<<(nz + 255) / 256, 256, 0, stream>>>(nbr_sum, nz);

    const int nth = N_EDGES * 32;   // one lane per (edge, column)
    edge_scatter_kernel<<<(nth + 255) / 256, 256, 0, stream>>>(
        intensity, edge_src, edge_dst, nbr_sum, deg, N_EDGES);

    fused_node_kernel<<<(N_NODES + 127) / 128, 256, 0, stream>>>(
        u, intensity, nbr_sum, deg, Wf, bf, Wg, bg, Wz, bz, WA, bA, out);
}